// AdaptiveFrequencyModulation_57294863729402
// MI455X (gfx1250) — compile-verified
//
#include <hip/hip_runtime.h>
#include <hip/hip_bf16.h>
#include <stdint.h>

// Phase-preserving style transfer for MI455X (gfx1250).
//  - detail bands: elementwise, LUT for cos(blended binary phase), float4 loads.
//  - approx band: 4-pass LSD radix sort (u32 keys = bit pattern of |x|, order-
//    preserving for non-negative floats) over 48 jobs (24 content-|.|, 24 style-|.|).
//    Cross-tile digit scan is computed as P = L_strict x H with V_WMMA_F32_16X16X4_F32.
//  - scatter kernel tile staging uses GLOBAL_LOAD_ASYNC_TO_LDS_B128 (ASYNCcnt path).

#define NSLICES   24
#define N_SLICE   262144          // 512*512 elements per (b,c) slice
#define NJOBS     48              // 24 source jobs + 24 target jobs
#define TILE      2048
#define TBLOCKS   128             // N_SLICE / TILE
#define NDIG      256
#define NTHREADS  256
#define EPT       8               // elements per thread per tile

typedef float v2f __attribute__((ext_vector_type(2)));
typedef float v8f __attribute__((ext_vector_type(8)));
typedef int   v4i __attribute__((vector_size(4 * sizeof(int))));
typedef __attribute__((address_space(1))) v4i* gv4i_p;   // global
typedef __attribute__((address_space(3))) v4i* lv4i_p;   // LDS

// cos(pi*(a*[c<0] + (1-a)*[s<0])), index = (c<0)<<1 | (s<0)
__constant__ float LUT_APPROX[4] = { 1.0f,  0.80901699437494745f, -0.80901699437494745f, -1.0f }; // a=0.8
__constant__ float LUT_DETAIL[4] = { 1.0f, -0.30901699437494745f,  0.30901699437494745f, -1.0f }; // a=0.4

__device__ __forceinline__ uint32_t key_of(float x) {
    return __float_as_uint(fabsf(x));       // |x| >= 0: IEEE bits sort like values
}

// ---------------------------------------------------------------- init keys
__global__ void pp_init_kernel(const float* __restrict__ content_approx,
                               const float* __restrict__ style_approx,
                               uint32_t* __restrict__ keys,
                               uint32_t* __restrict__ pays) {
    int job = blockIdx.y;
    int idx = blockIdx.x * NTHREADS + threadIdx.x;          // 0..N_SLICE-1
    const float* src = (job < NSLICES) ? (content_approx + (size_t)job * N_SLICE)
                                       : (style_approx + (size_t)(job - NSLICES) * N_SLICE);
    size_t o = (size_t)job * N_SLICE + idx;
    keys[o] = key_of(src[idx]);
    pays[o] = (uint32_t)idx;
}

// ---------------------------------------------------------------- per-tile histogram
__global__ void pp_hist_kernel(const uint32_t* __restrict__ keys,
                               uint32_t* __restrict__ hist, int shift) {
    __shared__ uint32_t bins[NDIG];
    int t = threadIdx.x, blk = blockIdx.x, job = blockIdx.y;
    bins[t] = 0;
    __syncthreads();
    size_t base = (size_t)job * N_SLICE + (size_t)blk * TILE;
    for (int j = 0; j < EPT; ++j) {
        uint32_t k = keys[base + (size_t)j * NTHREADS + t];
        atomicAdd(&bins[(k >> shift) & 255u], 1u);
    }
    __syncthreads();
    hist[((size_t)job * TBLOCKS + blk) * NDIG + t] = bins[t];
}

// ------------------------------------------------- cross-tile scan via WMMA matmul
// offs[b][d] = excl_scan_digits(total)[d] + sum_{b'<b} hist[b'][d]
// The second term is P = L_strict(128x128) * H(128x256), done with f32 WMMA 16x16x4.
__global__ void pp_scan_wmma_kernel(const uint32_t* __restrict__ hist,
                                    uint32_t* __restrict__ offs) {
    __shared__ float digitBase[NDIG];
    __shared__ float sc[NDIG];
    int t = threadIdx.x;
    int job = blockIdx.x;
    const uint32_t* H = hist + (size_t)job * TBLOCKS * NDIG;
    uint32_t* O = offs + (size_t)job * TBLOCKS * NDIG;

    // phase 1: per-digit totals, then exclusive scan across 256 digits
    float tot = 0.f;
    for (int k = 0; k < TBLOCKS; ++k) tot += (float)H[k * NDIG + t];
    sc[t] = tot;
    __syncthreads();
    for (int off = 1; off < NDIG; off <<= 1) {               // Hillis-Steele inclusive
        float v = (t >= off) ? sc[t - off] : 0.f;
        __syncthreads();
        sc[t] += v;
        __syncthreads();
    }
    digitBase[t] = sc[t] - tot;                              // exclusive
    __syncthreads();

    // phase 2: matmul scan. 8 waves, 128 output tiles of 16x16, K=128 in chunks of 4.
    int lane = t & 31, wave = t >> 5;
    int half = lane >> 4, l16 = lane & 15;
    for (int tile = wave; tile < (TBLOCKS / 16) * (NDIG / 16); tile += NTHREADS / 32) {
        int mT = tile >> 4, nT = tile & 15;
        int n = nT * 16 + l16;       // digit column (B/C/D lane mapping)
        int m = mT * 16 + l16;       // output row   (A lane mapping)
        v8f c = {};
        for (int kc = 0; kc < TBLOCKS / 4; ++kc) {
            int k0 = kc * 4 + half * 2;                      // VGPR0/VGPR1 K-pairing
            v2f a, b;
            a.x = ((float)(k0     < m));                     // strictly-lower triangular
            a.y = ((float)(k0 + 1 < m));
            b.x = (float)H[(k0    ) * NDIG + n];
            b.y = (float)H[(k0 + 1) * NDIG + n];
            c = __builtin_amdgcn_wmma_f32_16x16x4_f32(
                    /*neg_a=*/false, a, /*neg_b=*/false, b,
                    /*c_mod=*/(short)0, c, /*reuse_a=*/false, /*reuse_b=*/false);
        }
        #pragma unroll
        for (int vv = 0; vv < 8; ++vv) {                     // C/D layout: M = vv + 8*half
            int row = mT * 16 + vv + 8 * half;
            O[row * NDIG + n] = (uint32_t)(digitBase[n] + c[vv]);
        }
    }
}

// ---------------------------------------- stable intra-tile sort (8x 1-bit split) + scatter
__global__ void pp_scatter_kernel(const uint32_t* __restrict__ keysIn,
                                  const uint32_t* __restrict__ paysIn,
                                  uint32_t* __restrict__ keysOut,
                                  uint32_t* __restrict__ paysOut,
                                  const uint32_t* __restrict__ offs, int shift) {
    __shared__ uint32_t kbuf[2][TILE];
    __shared__ uint32_t pbuf[2][TILE];
    __shared__ uint32_t s1[NTHREADS];
    __shared__ uint32_t tileStart[NDIG];
    __shared__ uint32_t gbase[NDIG];

    int t = threadIdx.x, blk = blockIdx.x, job = blockIdx.y;
    size_t jobBase  = (size_t)job * N_SLICE;
    size_t tileBase = jobBase + (size_t)blk * TILE;

#if __has_builtin(__builtin_amdgcn_global_load_async_to_lds_b128)
    // CDNA5 async global->LDS copy (ASYNCcnt): stage the tile without VGPR round-trip.
    for (int j = 0; j < TILE / (NTHREADS * 4); ++j) {        // 2 iters, 16B per lane
        int p = (j * NTHREADS + t) * 4;
        __builtin_amdgcn_global_load_async_to_lds_b128(
            (gv4i_p)(keysIn + tileBase + p), (lv4i_p)&kbuf[0][p], 0, 0);
        __builtin_amdgcn_global_load_async_to_lds_b128(
            (gv4i_p)(paysIn + tileBase + p), (lv4i_p)&pbuf[0][p], 0, 0);
    }
    gbase[t] = offs[((size_t)job * TBLOCKS + blk) * NDIG + t];
  #if __has_builtin(__builtin_amdgcn_s_wait_asynccnt)
    __builtin_amdgcn_s_wait_asynccnt(0);
  #else
    asm volatile("s_wait_asynccnt 0" ::: "memory");
  #endif
#else
    gbase[t] = offs[((size_t)job * TBLOCKS + blk) * NDIG + t];
    for (int j = 0; j < EPT; ++j) {
        int p = j * NTHREADS + t;                            // coalesced load
        kbuf[0][p] = keysIn[tileBase + p];
        pbuf[0][p] = paysIn[tileBase + p];
    }
#endif
    __syncthreads();

    int cur = 0;
    for (int r = 0; r < 8; ++r) {
        uint32_t bit = 1u << (shift + r);
        uint32_t k[EPT], pay[EPT], pre[EPT];
        uint32_t cnt = 0;
        #pragma unroll
        for (int j = 0; j < EPT; ++j) {                      // contiguous chunk => stable
            int p = t * EPT + j;
            k[j]   = kbuf[cur][p];
            pay[j] = pbuf[cur][p];
            pre[j] = cnt;
            cnt += (k[j] & bit) ? 1u : 0u;
        }
        s1[t] = cnt;
        __syncthreads();
        for (int off = 1; off < NTHREADS; off <<= 1) {
            uint32_t v = (t >= off) ? s1[t - off] : 0u;
            __syncthreads();
            s1[t] += v;
            __syncthreads();
        }
        uint32_t excl       = s1[t] - cnt;
        uint32_t totalOnes  = s1[NTHREADS - 1];
        uint32_t totalZeros = TILE - totalOnes;
        #pragma unroll
        for (int j = 0; j < EPT; ++j) {
            int p = t * EPT + j;
            uint32_t onesBefore = excl + pre[j];
            uint32_t np = (k[j] & bit) ? (totalZeros + onesBefore)
                                       : ((uint32_t)p - onesBefore);
            kbuf[cur ^ 1][np] = k[j];
            pbuf[cur ^ 1][np] = pay[j];
        }
        __syncthreads();
        cur ^= 1;
    }

    // digit segment starts by boundary detection on the sorted tile
    for (int j = 0; j < EPT; ++j) {
        int p = j * NTHREADS + t;
        uint32_t d = (kbuf[cur][p] >> shift) & 255u;
        if (p == 0 || (((kbuf[cur][p - 1] >> shift) & 255u) != d))
            tileStart[d] = (uint32_t)p;
    }
    __syncthreads();

    for (int j = 0; j < EPT; ++j) {
        int p = j * NTHREADS + t;
        uint32_t key = kbuf[cur][p];
        uint32_t d   = (key >> shift) & 255u;
        uint32_t dst = gbase[d] + ((uint32_t)p - tileStart[d]);
        keysOut[jobBase + dst] = key;
        paysOut[jobBase + dst] = pbuf[cur][p];
    }
}

// ---------------------------------------------------------------- approx apply
// out[order[j]] = style_sorted[j] * cosLUT(signs at order[j])
__global__ void pp_apply_approx_kernel(const uint32_t* __restrict__ keys,
                                       const uint32_t* __restrict__ pays,
                                       const float* __restrict__ content_approx,
                                       const float* __restrict__ style_approx,
                                       float* __restrict__ out) {
    int s = blockIdx.y;
    int j = blockIdx.x * NTHREADS + threadIdx.x;
    uint32_t i = pays[(size_t)s * N_SLICE + j];                        // source argsort payload
    float mag  = __uint_as_float(keys[(size_t)(NSLICES + s) * N_SLICE + j]); // sorted |style|
    size_t gi = (size_t)s * N_SLICE + i;
    float c  = content_approx[gi];
    float st = style_approx[gi];
    int lidx = ((c < 0.f) ? 2 : 0) | ((st < 0.f) ? 1 : 0);
    out[gi] = mag * LUT_APPROX[lidx];
}

// ---------------------------------------------------------------- detail bands
__device__ __forceinline__ float blend_det(float a, float b) {
    int lidx = ((a < 0.f) ? 2 : 0) | ((b < 0.f) ? 1 : 0);
    return (0.4f * fabsf(a) + 0.6f * fabsf(b)) * LUT_DETAIL[lidx];
}

__global__ void pp_detail_kernel(const float4* __restrict__ ch, const float4* __restrict__ sh,
                                 const float4* __restrict__ cv, const float4* __restrict__ sv,
                                 const float4* __restrict__ cd, const float4* __restrict__ sd,
                                 float4* __restrict__ out4) {
    int band = blockIdx.y;
    size_t i = (size_t)blockIdx.x * NTHREADS + threadIdx.x;            // over N_TOTAL/4
    const float4* c = (band == 0) ? ch : (band == 1) ? cv : cd;
    const float4* s = (band == 0) ? sh : (band == 1) ? sv : sd;
    __builtin_prefetch(&c[i + 2048], 0, 1);                            // global_prefetch_b8
    __builtin_prefetch(&s[i + 2048], 0, 1);
    float4 a = c[i], b = s[i];
    float4 r;
    r.x = blend_det(a.x, b.x);
    r.y = blend_det(a.y, b.y);
    r.z = blend_det(a.z, b.z);
    r.w = blend_det(a.w, b.w);
    out4[(size_t)(1 + band) * (NSLICES * (size_t)N_SLICE / 4) + i] = r;
}

// ---------------------------------------------------------------- launch
extern "C" void kernel_launch(void* const* d_in, const int* in_sizes, int n_in,
                              void* d_out, int out_size, void* d_ws, size_t ws_size,
                              hipStream_t stream) {
    (void)in_sizes; (void)n_in; (void)out_size; (void)ws_size;
    const float* content_approx = (const float*)d_in[0];
    const float* content_h      = (const float*)d_in[1];
    const float* content_v      = (const float*)d_in[2];
    const float* content_d      = (const float*)d_in[3];
    const float* style_approx   = (const float*)d_in[4];
    const float* style_h        = (const float*)d_in[5];
    const float* style_v        = (const float*)d_in[6];
    const float* style_d        = (const float*)d_in[7];
    float* out = (float*)d_out;

    char* ws = (char*)d_ws;
    size_t keyBytes = (size_t)NJOBS * N_SLICE * 4;
    uint32_t* keysA = (uint32_t*)ws;  ws += keyBytes;
    uint32_t* keysB = (uint32_t*)ws;  ws += keyBytes;
    uint32_t* paysA = (uint32_t*)ws;  ws += keyBytes;
    uint32_t* paysB = (uint32_t*)ws;  ws += keyBytes;
    uint32_t* hist  = (uint32_t*)ws;  ws += (size_t)NJOBS * TBLOCKS * NDIG * 4;
    uint32_t* offs  = (uint32_t*)ws;

    pp_init_kernel<<<dim3(N_SLICE / NTHREADS, NJOBS), NTHREADS, 0, stream>>>(
        content_approx, style_approx, keysA, paysA);

    uint32_t *kc = keysA, *pc = paysA, *kn = keysB, *pn = paysB;
    for (int pass = 0; pass < 4; ++pass) {
        int shift = pass * 8;
        pp_hist_kernel<<<dim3(TBLOCKS, NJOBS), NTHREADS, 0, stream>>>(kc, hist, shift);
        pp_scan_wmma_kernel<<<dim3(NJOBS), NTHREADS, 0, stream>>>(hist, offs);
        pp_scatter_kernel<<<dim3(TBLOCKS, NJOBS), NTHREADS, 0, stream>>>(kc, pc, kn, pn, offs, shift);
        uint32_t* tk = kc; kc = kn; kn = tk;
        uint32_t* tp = pc; pc = pn; pn = tp;
    }

    pp_apply_approx_kernel<<<dim3(N_SLICE / NTHREADS, NSLICES), NTHREADS, 0, stream>>>(
        kc, pc, content_approx, style_approx, out);

    pp_detail_kernel<<<dim3((NSLICES * (N_SLICE / 4)) / NTHREADS, 3), NTHREADS, 0, stream>>>(
        (const float4*)content_h, (const float4*)style_h,
        (const float4*)content_v, (const float4*)style_v,
        (const float4*)content_d, (const float4*)style_d,
        (float4*)d_out);
}